// GCN_1975684956587
// MI455X (gfx1250) — compile-verified
//
#include <hip/hip_runtime.h>

typedef __attribute__((ext_vector_type(2))) float v2f;
typedef __attribute__((ext_vector_type(8))) float v8f;

#define N_NODES 100000
#define N_EDGES 1600000
#define IN_DIM  512
#define HID     32
#define OUT_DIM 40

// ---------------------------------------------------------------- degree prep
__global__ void k_init_deg(float* __restrict__ deg) {
    int i = blockIdx.x * blockDim.x + threadIdx.x;
    if (i < N_NODES) deg[i] = 1.0f;  // self-loop counted
}

__global__ void k_count(const int* __restrict__ dst, float* __restrict__ deg) {
    int e = blockIdx.x * blockDim.x + threadIdx.x;
    if (e < N_EDGES) atomicAdd(&deg[dst[e]], 1.0f);
}

__global__ void k_dinv(float* __restrict__ deg) {
    int i = blockIdx.x * blockDim.x + threadIdx.x;
    if (i < N_NODES) deg[i] = __frsqrt_rn(deg[i]);  // deg >= 1 always
}

// ---------------------------------------------------------------- GEMM1 (WMMA)
// h = x @ W1 (100000x512 @ 512x32), per-wave 16x32 output tile.
// W1 is pre-swizzled in LDS into per-lane B fragments so each k-step does one
// ds_load_b64 per column tile (no VGPR repacking before v_wmma).
// Fused: out1 = h * dinv[row]^2 (self-loop term of the propagation).
__global__ void __launch_bounds__(256)
k_gemm1(const float* __restrict__ x, const float* __restrict__ W1,
        const float* __restrict__ dinv,
        float* __restrict__ h, float* __restrict__ out1) {
    // fragment layout: [k4][ct][lane] -> v2f, k4 = k/4, ct = column tile (0,1)
    __shared__ v2f lwf[(IN_DIM / 4) * 2 * 32];  // 64 KB
    for (int idx = threadIdx.x; idx < (IN_DIM / 4) * 2 * 32; idx += 256) {
        const int ln  = idx & 31;
        const int ct  = (idx >> 5) & 1;
        const int k4  = idx >> 6;
        const int kb  = k4 * 4 + ((ln >> 4) << 1);   // lanes 16..31 hold K+2,K+3
        const int col = (ln & 15) + ct * 16;
        v2f v;
        v.x = W1[kb * HID + col];
        v.y = W1[(kb + 1) * HID + col];
        lwf[idx] = v;
    }
    __syncthreads();

    const int wave = threadIdx.x >> 5;
    const int lane = threadIdx.x & 31;
    const int tile = blockIdx.x * 8 + wave;     // one 16-row tile per wave
    if (tile * 16 >= N_NODES) return;           // wave-uniform exit (N%16==0)
    const int m0    = tile * 16;
    const int lrow  = lane & 15;
    const int khalf = (lane >> 4) * 2;

    const float* xrow  = x + (size_t)(m0 + lrow) * IN_DIM + khalf;
    const v2f*   bfrag = lwf + lane;

    v8f c0 = {}; v8f c1 = {};
#pragma unroll 4
    for (int k4 = 0; k4 < IN_DIM / 4; ++k4) {
        v2f a  = *(const v2f*)(xrow + k4 * 4);   // A 16x4 fragment (b64 load)
        v2f b0 = bfrag[k4 * 64];                 // cols 0..15  (ds_load_b64)
        v2f b1 = bfrag[k4 * 64 + 32];            // cols 16..31
        c0 = __builtin_amdgcn_wmma_f32_16x16x4_f32(false, a, false, b0, (short)0, c0, false, false);
        c1 = __builtin_amdgcn_wmma_f32_16x16x4_f32(false, a, false, b1, (short)0, c1, false, false);
    }

    // C/D layout: VGPR r -> row m0 + (lane>=16 ? 8 : 0) + r, col = lane%16
    const int rbase = m0 + ((lane >> 4) << 3);
#pragma unroll
    for (int r = 0; r < 8; ++r) {
        const int row = rbase + r;
        const float di = dinv[row];
        const float w  = di * di;
        const size_t o = (size_t)row * HID + lrow;
        h[o]         = c0[r];
        h[o + 16]    = c1[r];
        out1[o]      = c0[r] * w;
        out1[o + 16] = c1[r] * w;
    }
}

// ---------------------------------------------------------------- edge scatter L1
// out1[d,:] += h[s,:] * dinv[s]*dinv[d]; 8 threads/edge, 4 cols each.
__global__ void k_scatter1(const int* __restrict__ src, const int* __restrict__ dst,
                           const float* __restrict__ dinv,
                           const float* __restrict__ h, float* __restrict__ out1) {
    long long t = (long long)blockIdx.x * blockDim.x + threadIdx.x;
    int e = (int)(t >> 3);
    if (e >= N_EDGES) return;
    int cb = ((int)t & 7) * 4;
    int s = src[e], d = dst[e];
    float w = dinv[s] * dinv[d];
    float4 hv = *(const float4*)(h + (size_t)s * HID + cb);
    float* op = out1 + (size_t)d * HID + cb;
    atomicAdd(op + 0, hv.x * w);
    atomicAdd(op + 1, hv.y * w);
    atomicAdd(op + 2, hv.z * w);
    atomicAdd(op + 3, hv.w * w);
}

// ---------------------------------------------------------------- bias + relu
__global__ void k_relu(float* __restrict__ out1, const float* __restrict__ b1) {
    int i = blockIdx.x * blockDim.x + threadIdx.x;
    if (i < N_NODES * HID) {
        float v = out1[i] + b1[i & (HID - 1)];
        out1[i] = v > 0.0f ? v : 0.0f;
    }
}

// ---------------------------------------------------------------- GEMM2 (WMMA)
// h2 = h1 @ W2 (100000x32 @ 32x40). N padded to 3 column tiles of 16.
// Fused: out = h2 * dinv^2 + b2 (self-loop + bias init of final output).
__global__ void __launch_bounds__(256)
k_gemm2(const float* __restrict__ h1, const float* __restrict__ W2,
        const float* __restrict__ dinv, const float* __restrict__ b2,
        float* __restrict__ h2, float* __restrict__ out) {
    // fragment layout: [k4][ct][lane] -> v2f, ct in {0,1,2}; cols >= 40 are zero
    __shared__ v2f lwf[(HID / 4) * 3 * 32];  // 6 KB
    for (int idx = threadIdx.x; idx < (HID / 4) * 3 * 32; idx += 256) {
        const int ln  = idx & 31;
        const int ct  = (idx >> 5) % 3;
        const int k4  = idx / 96;
        const int kb  = k4 * 4 + ((ln >> 4) << 1);
        const int col = (ln & 15) + ct * 16;
        v2f v;
        v.x = (col < OUT_DIM) ? W2[kb * OUT_DIM + col] : 0.0f;
        v.y = (col < OUT_DIM) ? W2[(kb + 1) * OUT_DIM + col] : 0.0f;
        lwf[idx] = v;
    }
    __syncthreads();

    const int wave = threadIdx.x >> 5;
    const int lane = threadIdx.x & 31;
    const int tile = blockIdx.x * 8 + wave;
    if (tile * 16 >= N_NODES) return;
    const int m0    = tile * 16;
    const int lrow  = lane & 15;
    const int khalf = (lane >> 4) * 2;

    const float* hrow  = h1 + (size_t)(m0 + lrow) * HID + khalf;
    const v2f*   bfrag = lwf + lane;

    v8f c0 = {}; v8f c1 = {}; v8f c2 = {};
#pragma unroll
    for (int k4 = 0; k4 < HID / 4; ++k4) {
        v2f a  = *(const v2f*)(hrow + k4 * 4);
        v2f b0 = bfrag[k4 * 96];
        v2f b1 = bfrag[k4 * 96 + 32];
        v2f b2v = bfrag[k4 * 96 + 64];
        c0 = __builtin_amdgcn_wmma_f32_16x16x4_f32(false, a, false, b0,  (short)0, c0, false, false);
        c1 = __builtin_amdgcn_wmma_f32_16x16x4_f32(false, a, false, b1,  (short)0, c1, false, false);
        c2 = __builtin_amdgcn_wmma_f32_16x16x4_f32(false, a, false, b2v, (short)0, c2, false, false);
    }

    const int rbase = m0 + ((lane >> 4) << 3);
#pragma unroll
    for (int r = 0; r < 8; ++r) {
        const int row = rbase + r;
        const float di = dinv[row];
        const float w  = di * di;
        float* hr  = h2  + (size_t)row * OUT_DIM;
        float* orw = out + (size_t)row * OUT_DIM;
        hr[lrow]       = c0[r];
        orw[lrow]      = c0[r] * w + b2[lrow];
        hr[lrow + 16]  = c1[r];
        orw[lrow + 16] = c1[r] * w + b2[lrow + 16];
        if (lrow < OUT_DIM - 32) {               // cols 32..39 only
            hr[lrow + 32]  = c2[r];
            orw[lrow + 32] = c2[r] * w + b2[lrow + 32];
        }
    }
}

// ---------------------------------------------------------------- edge scatter L2
__global__ void k_scatter2(const int* __restrict__ src, const int* __restrict__ dst,
                           const float* __restrict__ dinv,
                           const float* __restrict__ h2, float* __restrict__ out) {
    long long t = (long long)blockIdx.x * blockDim.x + threadIdx.x;
    int e = (int)(t / 10);
    if (e >= N_EDGES) return;
    int cb = (int)(t % 10) * 4;
    int s = src[e], d = dst[e];
    float w = dinv[s] * dinv[d];
    float4 hv = *(const float4*)(h2 + (size_t)s * OUT_DIM + cb);
    float* op = out + (size_t)d * OUT_DIM + cb;
    atomicAdd(op + 0, hv.x * w);
    atomicAdd(op + 1, hv.y * w);
    atomicAdd(op + 2, hv.z * w);
    atomicAdd(op + 3, hv.w * w);
}

// ---------------------------------------------------------------- launcher
static inline size_t alignup(size_t v, size_t a) { return (v + a - 1) & ~(a - 1); }

extern "C" void kernel_launch(void* const* d_in, const int* in_sizes, int n_in,
                              void* d_out, int out_size, void* d_ws, size_t ws_size,
                              hipStream_t stream) {
    (void)in_sizes; (void)n_in; (void)out_size; (void)ws_size;
    const float* x   = (const float*)d_in[0];
    const int*   ei  = (const int*)d_in[1];      // [2, E] row-major
    const float* W1  = (const float*)d_in[2];
    const float* b1  = (const float*)d_in[3];
    const float* W2  = (const float*)d_in[4];
    const float* b2  = (const float*)d_in[5];
    float* out = (float*)d_out;

    const int* src = ei;
    const int* dst = ei + N_EDGES;

    // workspace carve-up
    size_t off = 0;
    float* dinv = (float*)((char*)d_ws + off); off = alignup(off + (size_t)N_NODES * 4, 256);
    float* h    = (float*)((char*)d_ws + off); off = alignup(off + (size_t)N_NODES * HID * 4, 256);
    float* out1 = (float*)((char*)d_ws + off); off = alignup(off + (size_t)N_NODES * HID * 4, 256);
    float* h2   = (float*)((char*)d_ws + off); off = alignup(off + (size_t)N_NODES * OUT_DIM * 4, 256);

    const int B = 256;
    const int gN   = (N_NODES + B - 1) / B;
    const int gE   = (N_EDGES + B - 1) / B;
    const int gRow = ((N_NODES / 16) + 7) / 8;   // 8 waves (tiles) per block
    const int gS1  = (int)(((long long)N_EDGES * 8  + B - 1) / B);
    const int gS2  = (int)(((long long)N_EDGES * 10 + B - 1) / B);
    const int gHW  = (N_NODES * HID + B - 1) / B;

    k_init_deg<<<gN, B, 0, stream>>>(dinv);
    k_count   <<<gE, B, 0, stream>>>(dst, dinv);
    k_dinv    <<<gN, B, 0, stream>>>(dinv);

    k_gemm1   <<<gRow, B, 0, stream>>>(x, W1, dinv, h, out1);
    k_scatter1<<<gS1, B, 0, stream>>>(src, dst, dinv, h, out1);
    k_relu    <<<gHW, B, 0, stream>>>(out1, b1);

    k_gemm2   <<<gRow, B, 0, stream>>>(out1, W2, dinv, b2, h2, out);
    k_scatter2<<<gS2, B, 0, stream>>>(src, dst, dinv, h2, out);
}